// BPPGraphEncoder_24601572671728
// MI455X (gfx1250) — compile-verified
//
#include <hip/hip_runtime.h>
#include <hip/hip_bf16.h>

#define N_NODES 50000
#define DIM     128
#define N_EDGES 1600000
#define ND      (N_NODES * DIM)       // 6,400,000
#define NPART   512

typedef __attribute__((ext_vector_type(2))) float v2f;
typedef __attribute__((ext_vector_type(8))) float v8f;

// ---------------------------------------------------------------------------
// QKV projection: out = x @ W^T + b for one of {Q,K,V} (blockIdx.y selects).
// One block = 16-row strip of x staged in LDS; 8 waves, one 16-col tile each.
// Uses V_WMMA_F32_16X16X4_F32, K-loop of 32 steps.
// ---------------------------------------------------------------------------
__global__ __launch_bounds__(256)
void qkv_gemm_kernel(const float* __restrict__ x,
                     const float* __restrict__ Wq, const float* __restrict__ bq,
                     const float* __restrict__ Wk, const float* __restrict__ bk,
                     const float* __restrict__ Wv, const float* __restrict__ bv,
                     float* __restrict__ Q, float* __restrict__ K,
                     float* __restrict__ V)
{
    __shared__ float xs[16 * DIM];   // 8 KB strip of x
    const int tid  = threadIdx.x;
    const int row0 = blockIdx.x * 16;

    const float* W;  const float* bias;  float* out;
    if      (blockIdx.y == 0) { W = Wq; bias = bq; out = Q; }
    else if (blockIdx.y == 1) { W = Wk; bias = bk; out = K; }
    else                      { W = Wv; bias = bv; out = V; }

    // Cooperative load of x[row0:row0+16, 0:128] (contiguous 2048 floats).
    for (int t = tid; t < 512; t += 256) {
        *(float4*)(xs + t * 4) = *(const float4*)(x + row0 * DIM + t * 4);
    }
    __syncthreads();

    const int lane = tid & 31;
    const int col0 = (tid >> 5) * 16;     // wave id * 16
    const int half = lane >> 4;           // 0: K pair {0,1}, 1: K pair {2,3}
    const int l16  = lane & 15;

    v8f acc = {};
    #pragma unroll 4
    for (int k = 0; k < DIM; k += 4) {
        // A: 16x4 f32 fragment from LDS (row l16, K base k + 2*half)
        v2f a = *(const v2f*)(xs + l16 * DIM + k + half * 2);
        // B: 4x16 fragment; B(kk,jj) = W[(col0+jj)*DIM + k+kk]
        v2f b = *(const v2f*)(W + (col0 + l16) * DIM + k + half * 2);
        acc = __builtin_amdgcn_wmma_f32_16x16x4_f32(
            /*neg_a=*/false, a, /*neg_b=*/false, b,
            /*c_mod=*/(short)0, acc, /*reuse_a=*/false, /*reuse_b=*/false);
    }

    // C/D layout: VGPR r -> M = r + 8*half, N = l16
    const float bb = bias[col0 + l16];
    const int mrow = row0 + half * 8;
    #pragma unroll
    for (int r = 0; r < 8; ++r) {
        out[(mrow + r) * DIM + col0 + l16] = acc[r] + bb;
    }
}

// ---------------------------------------------------------------------------
// Edge scores: s[e] = dot(Q[row[e]], K[col[e]]) / sqrt(DIM).
// One wave (32 lanes x float4) per edge; 8 edges per block.
// ---------------------------------------------------------------------------
__global__ __launch_bounds__(256)
void edge_scores_kernel(const float* __restrict__ Q, const float* __restrict__ K,
                        const int* __restrict__ row, const int* __restrict__ col,
                        float* __restrict__ s)
{
    const int e    = blockIdx.x * 8 + (threadIdx.x >> 5);
    const int lane = threadIdx.x & 31;
    const int r = row[e], c = col[e];
    const float4 q = *(const float4*)(Q + r * DIM + lane * 4);
    const float4 k = *(const float4*)(K + c * DIM + lane * 4);
    float p = q.x * k.x + q.y * k.y + q.z * k.z + q.w * k.w;
    #pragma unroll
    for (int off = 16; off > 0; off >>= 1) p += __shfl_xor(p, off, 32);
    if (lane == 0) s[e] = p * 0.08838834764831845f;   // 1/sqrt(128)
}

// ---------------------------------------------------------------------------
// Deterministic two-stage reductions for global softmax (max, then exp-sum).
// ---------------------------------------------------------------------------
__device__ __forceinline__ float block_reduce_max(float m) {
    __shared__ float sm[8];
    #pragma unroll
    for (int off = 16; off > 0; off >>= 1) m = fmaxf(m, __shfl_xor(m, off, 32));
    if ((threadIdx.x & 31) == 0) sm[threadIdx.x >> 5] = m;
    __syncthreads();
    if (threadIdx.x < 8) {
        m = sm[threadIdx.x];
        #pragma unroll
        for (int off = 4; off > 0; off >>= 1) m = fmaxf(m, __shfl_xor(m, off, 32));
    }
    return m;
}

__device__ __forceinline__ float block_reduce_sum(float a) {
    __shared__ float sm[8];
    #pragma unroll
    for (int off = 16; off > 0; off >>= 1) a += __shfl_xor(a, off, 32);
    if ((threadIdx.x & 31) == 0) sm[threadIdx.x >> 5] = a;
    __syncthreads();
    if (threadIdx.x < 8) {
        a = sm[threadIdx.x];
        #pragma unroll
        for (int off = 4; off > 0; off >>= 1) a += __shfl_xor(a, off, 32);
    }
    return a;
}

__global__ __launch_bounds__(256)
void max_stage1_kernel(const float* __restrict__ s, float* __restrict__ partials) {
    float m = -3.402823466e38f;
    for (int i = blockIdx.x * 256 + threadIdx.x; i < N_EDGES; i += NPART * 256)
        m = fmaxf(m, s[i]);
    m = block_reduce_max(m);
    if (threadIdx.x == 0) partials[blockIdx.x] = m;
}

__global__ __launch_bounds__(256)
void max_stage2_kernel(const float* __restrict__ partials, float* __restrict__ out_m) {
    float m = fmaxf(partials[threadIdx.x], partials[threadIdx.x + 256]);
    m = block_reduce_max(m);
    if (threadIdx.x == 0) out_m[0] = m;
}

__global__ __launch_bounds__(256)
void expsum_stage1_kernel(const float* __restrict__ s, const float* __restrict__ mptr,
                          float* __restrict__ partials) {
    const float m = mptr[0];
    float a = 0.0f;
    for (int i = blockIdx.x * 256 + threadIdx.x; i < N_EDGES; i += NPART * 256)
        a += __expf(s[i] - m);
    a = block_reduce_sum(a);
    if (threadIdx.x == 0) partials[blockIdx.x] = a;
}

__global__ __launch_bounds__(256)
void sum_stage2_kernel(const float* __restrict__ partials, float* __restrict__ out_d) {
    float a = partials[threadIdx.x] + partials[threadIdx.x + 256];
    a = block_reduce_sum(a);
    if (threadIdx.x == 0) out_d[0] = a;
}

// ---------------------------------------------------------------------------
// Aggregation: agg[row[e]] += softmax(s)[e] * V[col[e]]   (native f32 atomics)
// One wave per edge, float4 per lane -> 4 global_atomic_add_f32 per lane.
// ---------------------------------------------------------------------------
__global__ __launch_bounds__(256)
void aggregate_kernel(const float* __restrict__ s, const float* __restrict__ V,
                      const int* __restrict__ row, const int* __restrict__ col,
                      const float* __restrict__ mptr, const float* __restrict__ dptr,
                      float* __restrict__ agg)
{
    const int e    = blockIdx.x * 8 + (threadIdx.x >> 5);
    const int lane = threadIdx.x & 31;
    const float w  = __expf(s[e] - mptr[0]) * (1.0f / dptr[0]);
    const int r = row[e], c = col[e];
    const float4 v = *(const float4*)(V + c * DIM + lane * 4);
    float* dst = agg + r * DIM + lane * 4;
    __hip_atomic_fetch_add(dst + 0, w * v.x, __ATOMIC_RELAXED, __HIP_MEMORY_SCOPE_AGENT);
    __hip_atomic_fetch_add(dst + 1, w * v.y, __ATOMIC_RELAXED, __HIP_MEMORY_SCOPE_AGENT);
    __hip_atomic_fetch_add(dst + 2, w * v.z, __ATOMIC_RELAXED, __HIP_MEMORY_SCOPE_AGENT);
    __hip_atomic_fetch_add(dst + 3, w * v.w, __ATOMIC_RELAXED, __HIP_MEMORY_SCOPE_AGENT);
}

// ---------------------------------------------------------------------------
// In-place ELU between layers.
// ---------------------------------------------------------------------------
__global__ __launch_bounds__(256)
void elu_kernel(float* __restrict__ h) {
    const int i = blockIdx.x * 256 + threadIdx.x;
    const float a = h[i];
    h[i] = (a > 0.0f) ? a : expm1f(a);
}

// ---------------------------------------------------------------------------
extern "C" void kernel_launch(void* const* d_in, const int* in_sizes, int n_in,
                              void* d_out, int out_size, void* d_ws, size_t ws_size,
                              hipStream_t stream) {
    const float* x    = (const float*)d_in[0];
    const int*   eidx = (const int*)d_in[1];
    const int*   row  = eidx;             // edge_index[0, :]
    const int*   col  = eidx + N_EDGES;   // edge_index[1, :]
    const float* Wq1 = (const float*)d_in[2];  const float* bq1 = (const float*)d_in[3];
    const float* Wk1 = (const float*)d_in[4];  const float* bk1 = (const float*)d_in[5];
    const float* Wv1 = (const float*)d_in[6];  const float* bv1 = (const float*)d_in[7];
    const float* Wq2 = (const float*)d_in[8];  const float* bq2 = (const float*)d_in[9];
    const float* Wk2 = (const float*)d_in[10]; const float* bk2 = (const float*)d_in[11];
    const float* Wv2 = (const float*)d_in[12]; const float* bv2 = (const float*)d_in[13];

    float* Q        = (float*)d_ws;
    float* K        = Q + ND;
    float* V        = K + ND;
    float* s        = V + ND;             // N_EDGES scores
    float* agg      = s + N_EDGES;        // layer-1 output / layer-2 input (ND)
    float* partials = agg + ND;           // NPART
    float* scal     = partials + NPART;   // [0]=max, [1]=denom

    const dim3 gG(N_NODES / 16, 3), bG(256);
    const int  gE = N_EDGES / 8;
    float* outp = (float*)d_out;

    // ---------------- Layer 1 ----------------
    qkv_gemm_kernel<<<gG, bG, 0, stream>>>(x, Wq1, bq1, Wk1, bk1, Wv1, bv1, Q, K, V);
    edge_scores_kernel<<<gE, 256, 0, stream>>>(Q, K, row, col, s);
    max_stage1_kernel<<<NPART, 256, 0, stream>>>(s, partials);
    max_stage2_kernel<<<1, 256, 0, stream>>>(partials, scal);
    expsum_stage1_kernel<<<NPART, 256, 0, stream>>>(s, scal, partials);
    sum_stage2_kernel<<<1, 256, 0, stream>>>(partials, scal + 1);
    hipMemsetAsync(agg, 0, (size_t)ND * sizeof(float), stream);
    aggregate_kernel<<<gE, 256, 0, stream>>>(s, V, row, col, scal, scal + 1, agg);
    elu_kernel<<<ND / 256, 256, 0, stream>>>(agg);

    // ---------------- Layer 2 ----------------
    qkv_gemm_kernel<<<gG, bG, 0, stream>>>(agg, Wq2, bq2, Wk2, bk2, Wv2, bv2, Q, K, V);
    edge_scores_kernel<<<gE, 256, 0, stream>>>(Q, K, row, col, s);
    max_stage1_kernel<<<NPART, 256, 0, stream>>>(s, partials);
    max_stage2_kernel<<<1, 256, 0, stream>>>(partials, scal);
    expsum_stage1_kernel<<<NPART, 256, 0, stream>>>(s, scal, partials);
    sum_stage2_kernel<<<1, 256, 0, stream>>>(partials, scal + 1);
    hipMemsetAsync(outp, 0, (size_t)ND * sizeof(float), stream);
    aggregate_kernel<<<gE, 256, 0, stream>>>(s, V, row, col, scal, scal + 1, outp);
}